// BackProjLayer_4604204941414
// MI455X (gfx1250) — compile-verified
//
#include <hip/hip_runtime.h>
#include <math.h>

typedef __attribute__((ext_vector_type(2))) float v2f;
typedef __attribute__((ext_vector_type(8))) float v8f;

#define NB   64
#define NCH  32
#define NPX  20000
#define KCH  25            // K-chunks per batch for the outer-product GEMM
#define TPB  50            // 16-pixel tiles per chunk (20000/16/25)
#define BETA 1.3130352854993313f

static __device__ __forceinline__ v8f wmma_f32(v2f a, v2f b, v8f c) {
  // V_WMMA_F32_16X16X4_F32 : D = A(16x4,f32) * B(4x16,f32) + C(16x16,f32)
  return __builtin_amdgcn_wmma_f32_16x16x4_f32(false, a, false, b, (short)0, c,
                                               false, false);
}

// LDS byte offset of a shared-memory pointer (AS(3) pointers are offsets
// from LDS_BASE on amdgcn).
static __device__ __forceinline__ unsigned lds_off(const void* p) {
  return (unsigned)(unsigned long long)
         (__attribute__((address_space(3))) const char*)p;
}

// ---------------------------------------------------------------------------
// Kernel 1: batched complex Hermitian eigh via parallel (tournament) Jacobi.
// One block per batch matrix; matrix + eigenvectors live in LDS.
// Writes Vsc = Vs * sqrt(max(eig,0)) as separate re/im planes into ws.
// ---------------------------------------------------------------------------
static __device__ __forceinline__ void jacobi_pair(int r, int k, int& p, int& q) {
  int a, b;
  if (k == 0) { a = 31; b = r % 31; }
  else        { a = (r + k) % 31; b = (r + 31 - k) % 31; }
  p = a < b ? a : b;
  q = a < b ? b : a;
}

__global__ __launch_bounds__(128)
void eigh_kernel(const float* __restrict__ Sre, const float* __restrict__ Sim,
                 float* __restrict__ Vre_out, float* __restrict__ Vim_out) {
  __shared__ float Ar[32][32], Ai[32][32], Vr[32][32], Vi[32][32];
  __shared__ float rc[16], rs[16], rpr[16], rpi[16];
  const int b   = blockIdx.x;
  const int tid = threadIdx.x;
  const float* sr = Sre + b * 1024;   // (B,1,32,32) squeezed
  const float* si = Sim + b * 1024;

  for (int i = tid; i < 1024; i += 128) {
    int r = i >> 5, c = i & 31;
    Ar[r][c] = sr[i];
    Ai[r][c] = si[i];
    Vr[r][c] = (r == c) ? 1.f : 0.f;
    Vi[r][c] = 0.f;
  }
  __syncthreads();

  for (int sweep = 0; sweep < 8; ++sweep) {
    for (int round = 0; round < 31; ++round) {
      // --- compute 16 commuting complex rotations from current A ---
      if (tid < 16) {
        int p, q; jacobi_pair(round, tid, p, q);
        float app = Ar[p][p], aqq = Ar[q][q];
        float ar  = Ar[p][q], ai  = Ai[p][q];
        float r2  = ar * ar + ai * ai;
        float c = 1.f, s = 0.f, phr = 1.f, phi = 0.f;
        if (r2 > 1e-26f) {
          float rmag = sqrtf(r2);
          phr = ar / rmag;            // ph = e^{-i*phi(apq)}
          phi = -ai / rmag;
          float ta = (aqq - app) / (2.f * rmag);
          float t  = (ta >= 0.f ? 1.f : -1.f) / (fabsf(ta) + sqrtf(1.f + ta * ta));
          c = 1.f / sqrtf(1.f + t * t);
          s = t * c;
        }
        rc[tid] = c; rs[tid] = s; rpr[tid] = phr; rpi[tid] = phi;
      }
      __syncthreads();
      // --- column update A := A*J, and V := V*J (disjoint column pairs) ---
      for (int tk = tid; tk < 512; tk += 128) {
        int k = tk >> 5, i = tk & 31;
        int p, q; jacobi_pair(round, k, p, q);
        float c = rc[k], s = rs[k], phr = rpr[k], phi = rpi[k];
        float aqr = Ar[i][q], aqi = Ai[i][q];
        float zr = phr * aqr - phi * aqi, zi = phr * aqi + phi * aqr;
        float apr = Ar[i][p], api = Ai[i][p];
        Ar[i][p] = c * apr - s * zr;  Ai[i][p] = c * api - s * zi;
        Ar[i][q] = s * apr + c * zr;  Ai[i][q] = s * api + c * zi;
        float vqr = Vr[i][q], vqi = Vi[i][q];
        float wr = phr * vqr - phi * vqi, wi = phr * vqi + phi * vqr;
        float vpr = Vr[i][p], vpi = Vi[i][p];
        Vr[i][p] = c * vpr - s * wr;  Vi[i][p] = c * vpi - s * wi;
        Vr[i][q] = s * vpr + c * wr;  Vi[i][q] = s * vpi + c * wi;
      }
      __syncthreads();
      // --- row update A := J^H * A (conjugate phase, disjoint row pairs) ---
      for (int tk = tid; tk < 512; tk += 128) {
        int k = tk >> 5, j = tk & 31;
        int p, q; jacobi_pair(round, k, p, q);
        float c = rc[k], s = rs[k], phr = rpr[k], phi = -rpi[k];
        float aqr = Ar[q][j], aqi = Ai[q][j];
        float wr = phr * aqr - phi * aqi, wi = phr * aqi + phi * aqr;
        float apr = Ar[p][j], api = Ai[p][j];
        Ar[p][j] = c * apr - s * wr;  Ai[p][j] = c * api - s * wi;
        Ar[q][j] = s * apr + c * wr;  Ai[q][j] = s * api + c * wi;
      }
      __syncthreads();
    }
  }
  // scale eigenvector columns by sqrt(clamped eigenvalue)
  for (int i = tid; i < 1024; i += 128) {
    int r = i >> 5, c = i & 31;
    float ev = Ar[c][c];
    float sc = ev > 0.f ? sqrtf(ev) : 0.f;
    Vre_out[b * 1024 + i] = Vr[r][c] * sc;
    Vim_out[b * 1024 + i] = Vi[r][c] * sc;
  }
}

// ---------------------------------------------------------------------------
// Kernel 2: latent = ||D^H Vs||^2 per pixel, then ReTanh.  WMMA f32 16x16x4.
// One wave per 32-pixel group (two M-tiles sharing the B fragments, halving
// the Vs load issue rate): M=pixels, K=channels(32), N=eig cols(32).
// ---------------------------------------------------------------------------
__global__ __launch_bounds__(256)
void latent_kernel(const float* __restrict__ Dre, const float* __restrict__ Dim,
                   const float* __restrict__ tau,
                   const float* __restrict__ Vre, const float* __restrict__ Vim,
                   float* __restrict__ latent_out) {
  const int b    = blockIdx.y;
  const int w    = threadIdx.x >> 5;
  const int lane = threadIdx.x & 31;
  const int g    = blockIdx.x * 8 + w;       // 32-pixel group, 625 per batch
  const int p0   = g * 32;
  if (p0 >= NPX) return;                     // wave-uniform guard
  const int m16 = lane & 15;
  const int hi  = lane >> 4;

  // A fragments: A[m=pixel][k=channel] = D[k][p], two 16-pixel M-tiles
  v2f Afr[2][8], Afi[2][8];
#pragma unroll
  for (int mt = 0; mt < 2; ++mt) {
    int p = p0 + mt * 16 + m16;
#pragma unroll
    for (int kk = 0; kk < 8; ++kk) {
      int k0 = kk * 4 + hi * 2;
      Afr[mt][kk].x = Dre[k0 * NPX + p];   Afr[mt][kk].y = Dre[(k0 + 1) * NPX + p];
      Afi[mt][kk].x = Dim[k0 * NPX + p];   Afi[mt][kk].y = Dim[(k0 + 1) * NPX + p];
    }
  }
  const float* vr = Vre + b * 1024;
  const float* vi = Vim + b * 1024;

  float s[2][8];
#pragma unroll
  for (int mt = 0; mt < 2; ++mt)
#pragma unroll
    for (int r = 0; r < 8; ++r) s[mt][r] = 0.f;

#pragma unroll
  for (int h = 0; h < 2; ++h) {            // two 16-column halves of N=32
    v8f cre[2], cim[2];
#pragma unroll
    for (int mt = 0; mt < 2; ++mt) {
      cre[mt] = (v8f){0,0,0,0,0,0,0,0};
      cim[mt] = (v8f){0,0,0,0,0,0,0,0};
    }
    int n = h * 16 + m16;
#pragma unroll
    for (int kk = 0; kk < 8; ++kk) {
      int k0 = kk * 4 + hi * 2;
      v2f br, bi;
      br.x = vr[k0 * 32 + n];  br.y = vr[(k0 + 1) * 32 + n];
      bi.x = vi[k0 * 32 + n];  bi.y = vi[(k0 + 1) * 32 + n];
#pragma unroll
      for (int mt = 0; mt < 2; ++mt) {
        v2f afin = -Afi[mt][kk];
        // conj(D)^T V : re = Dr*Vr + Di*Vi ; im = Dr*Vi - Di*Vr
        cre[mt] = wmma_f32(Afr[mt][kk], br, cre[mt]);
        cre[mt] = wmma_f32(Afi[mt][kk], bi, cre[mt]);
        cim[mt] = wmma_f32(Afr[mt][kk], bi, cim[mt]);
        cim[mt] = wmma_f32(afin,        br, cim[mt]);
      }
    }
#pragma unroll
    for (int mt = 0; mt < 2; ++mt)
#pragma unroll
      for (int r = 0; r < 8; ++r)
        s[mt][r] += cre[mt][r] * cre[mt][r] + cim[mt][r] * cim[mt][r];
  }
  // reduce over N (16 lanes within each half; xor masks stay inside halves)
#pragma unroll
  for (int off = 1; off < 16; off <<= 1)
#pragma unroll
    for (int mt = 0; mt < 2; ++mt)
#pragma unroll
      for (int r = 0; r < 8; ++r)
        s[mt][r] += __shfl_xor(s[mt][r], off, 32);

  if (m16 == 0) {
#pragma unroll
    for (int mt = 0; mt < 2; ++mt)
#pragma unroll
      for (int r = 0; r < 8; ++r) {
        int pp = p0 + mt * 16 + hi * 8 + r;  // C layout: vgpr r -> M = r + hi*8
        float v = s[mt][r] - tau[pp];
        latent_out[b * NPX + pp] = fmaxf(1e-6f, BETA * tanhf(v));
      }
  }
}

// ---------------------------------------------------------------------------
// Kernel 3: out = (A*diag(lam)) @ A^H, K=20000 split in 25 chunks per batch.
// Per-wave LDS staging via CDNA5 async global->LDS B128 copies (ASYNCcnt),
// padded stride-20 rows -> conflict-free ds reads, WMMA f32 16x16x4,
// deterministic block reduction to partials.  No block barriers in the main
// loop: staging is wave-private, guarded only by s_wait_asynccnt/dscnt.
// ---------------------------------------------------------------------------
#define LSTR 20                      // padded LDS row stride (floats)
#define WAVE_SMEM (2 * 32 * LSTR + 16)   // Ar tile + Ai tile + lam tile

__global__ __launch_bounds__(256)
void outer_kernel(const float* __restrict__ Are, const float* __restrict__ Aim,
                  const float* __restrict__ latent, float* __restrict__ part) {
  __shared__ float smem[8 * WAVE_SMEM];   // 41,472 B; reused for reduction
  const int b     = blockIdx.y;
  const int chunk = blockIdx.x;
  const int w     = threadIdx.x >> 5;
  const int lane  = threadIdx.x & 31;
  const int m16   = lane & 15;
  const int hi    = lane >> 4;

  float* sAr  = smem + w * WAVE_SMEM;     // [32][LSTR]
  float* sAi  = sAr + 32 * LSTR;
  float* sLam = sAi + 32 * LSTR;
  const unsigned lAr = lds_off(&sAr[lane * LSTR]);
  const unsigned lAi = lds_off(&sAi[lane * LSTR]);

  v8f cre[2][2], cim[2][2];
#pragma unroll
  for (int mt = 0; mt < 2; ++mt)
#pragma unroll
    for (int nt = 0; nt < 2; ++nt) {
      cre[mt][nt] = (v8f){0,0,0,0,0,0,0,0};
      cim[mt][nt] = (v8f){0,0,0,0,0,0,0,0};
    }
  const float* lat = latent + b * NPX;

  for (int it = 0; it < 7; ++it) {        // ceil(50 tiles / 8 waves)
    int t = it * 8 + w;
    bool active = t < TPB;
    if (active) {
      // WAR guard: previous tile's ds reads must have completed before the
      // async engine overwrites this wave's staging area.
      asm volatile("s_wait_dscnt 0x0" ::: "memory");
      int pb = (chunk * TPB + t) * 16;    // 16-pixel K tile
      const float* gr = Are + lane * NPX + pb;
      const float* gi = Aim + lane * NPX + pb;
#pragma unroll
      for (int j = 0; j < 4; ++j) {
        asm volatile("global_load_async_to_lds_b128 %0, %1, off"
                     :: "v"(lAr + j * 16), "v"(gr + j * 4) : "memory");
        asm volatile("global_load_async_to_lds_b128 %0, %1, off"
                     :: "v"(lAi + j * 16), "v"(gi + j * 4) : "memory");
      }
      if (lane < 16) sLam[lane] = lat[pb + lane];
      // consume only after async copies + lam ds_store have landed
      asm volatile("s_wait_asynccnt 0x0" ::: "memory");
      asm volatile("s_wait_dscnt 0x0" ::: "memory");
#pragma unroll
      for (int kk = 0; kk < 4; ++kk) {    // K=16 per tile, 4 per WMMA
        int k0 = kk * 4 + hi * 2;
        float l0 = sLam[k0], l1 = sLam[k0 + 1];
        v2f fBr[2], fBi[2], fnBr[2], gAr2[2], gAi2[2];
#pragma unroll
        for (int mt = 0; mt < 2; ++mt) {
          int c = mt * 16 + m16;
          float ar0 = sAr[c * LSTR + k0], ar1 = sAr[c * LSTR + k0 + 1];
          float ai0 = sAi[c * LSTR + k0], ai1 = sAi[c * LSTR + k0 + 1];
          fBr[mt].x = ar0 * l0;  fBr[mt].y = ar1 * l1;   // (A*lam) rows
          fBi[mt].x = ai0 * l0;  fBi[mt].y = ai1 * l1;
          fnBr[mt]  = -fBr[mt];
          gAr2[mt].x = ar0;      gAr2[mt].y = ar1;       // A^T cols (B frag)
          gAi2[mt].x = ai0;      gAi2[mt].y = ai1;
        }
#pragma unroll
        for (int mt = 0; mt < 2; ++mt)
#pragma unroll
          for (int nt = 0; nt < 2; ++nt) {
            // out_re = Br*Ar^T + Bi*Ai^T ; out_im = Bi*Ar^T - Br*Ai^T
            cre[mt][nt] = wmma_f32(fBr[mt],  gAr2[nt], cre[mt][nt]);
            cre[mt][nt] = wmma_f32(fBi[mt],  gAi2[nt], cre[mt][nt]);
            cim[mt][nt] = wmma_f32(fBi[mt],  gAr2[nt], cim[mt][nt]);
            cim[mt][nt] = wmma_f32(fnBr[mt], gAi2[nt], cim[mt][nt]);
          }
      }
    }
  }
  // all waves done with their staging reads before smem is repurposed
  asm volatile("s_wait_dscnt 0x0" ::: "memory");
  __syncthreads();

  // ---- deterministic reduction across 8 waves (two rounds, alias smem) ----
  float* red = smem;                       // 4 * 2048 floats = 32 KB
  if (w >= 4) {
#pragma unroll
    for (int mt = 0; mt < 2; ++mt)
#pragma unroll
      for (int nt = 0; nt < 2; ++nt)
#pragma unroll
        for (int r = 0; r < 8; ++r) {
          int M = mt * 16 + hi * 8 + r;
          int N = nt * 16 + m16;
          red[(w - 4) * 2048 +        M * 32 + N] = cre[mt][nt][r];
          red[(w - 4) * 2048 + 1024 + M * 32 + N] = cim[mt][nt][r];
        }
  }
  __syncthreads();
  if (w < 4) {
#pragma unroll
    for (int mt = 0; mt < 2; ++mt)
#pragma unroll
      for (int nt = 0; nt < 2; ++nt)
#pragma unroll
        for (int r = 0; r < 8; ++r) {
          int M = mt * 16 + hi * 8 + r;
          int N = nt * 16 + m16;
          red[w * 2048 +        M * 32 + N] += cre[mt][nt][r];
          red[w * 2048 + 1024 + M * 32 + N] += cim[mt][nt][r];
        }
  }
  __syncthreads();
  float* dst = part + (b * KCH + chunk) * 2048;
  for (int i = threadIdx.x; i < 2048; i += 256)
    dst[i] = red[i] + red[2048 + i] + red[4096 + i] + red[6144 + i];
}

// ---------------------------------------------------------------------------
// Kernel 4: reduce K-chunk partials -> interleaved complex output (B,32,32).
// ---------------------------------------------------------------------------
__global__ __launch_bounds__(256)
void reduce_kernel(const float* __restrict__ part, float* __restrict__ out) {
  const int b = blockIdx.x;
  for (int i = threadIdx.x; i < 2048; i += 256) {
    float acc = 0.f;
#pragma unroll
    for (int k = 0; k < KCH; ++k) acc += part[(b * KCH + k) * 2048 + i];
    int plane = i >> 10;        // 0 = re, 1 = im
    int cd    = i & 1023;
    out[(b * 1024 + cd) * 2 + plane] = acc;
  }
}

// ---------------------------------------------------------------------------
extern "C" void kernel_launch(void* const* d_in, const int* in_sizes, int n_in,
                              void* d_out, int out_size, void* d_ws, size_t ws_size,
                              hipStream_t stream) {
  (void)in_sizes; (void)n_in; (void)out_size; (void)ws_size;
  const float* Sre = (const float*)d_in[0];
  const float* Sim = (const float*)d_in[1];
  const float* tau = (const float*)d_in[2];
  const float* Dre = (const float*)d_in[3];
  const float* Dim = (const float*)d_in[4];
  const float* Are = (const float*)d_in[5];
  const float* Aim = (const float*)d_in[6];

  float* out    = (float*)d_out;
  float* latent = out + NB * NCH * NCH * 2;   // tuple: (out complex, latent)

  float* wsf  = (float*)d_ws;
  float* Vre  = wsf;                          // NB*1024
  float* Vim  = Vre + NB * 1024;              // NB*1024
  float* part = Vim + NB * 1024;              // NB*KCH*2048 (~13 MB)

  eigh_kernel<<<dim3(NB), dim3(128), 0, stream>>>(Sre, Sim, Vre, Vim);
  latent_kernel<<<dim3(79, NB), dim3(256), 0, stream>>>(Dre, Dim, tau, Vre, Vim, latent);
  outer_kernel<<<dim3(KCH, NB), dim3(256), 0, stream>>>(Are, Aim, latent, part);
  reduce_kernel<<<dim3(NB), dim3(256), 0, stream>>>(part, out);
}